// ObjectDetectionLoss_49177375539626
// MI455X (gfx1250) — compile-verified
//
#include <hip/hip_runtime.h>
#include <stdint.h>

// ---------------- tunables / problem constants ----------------
#define BATCH   32
#define NANCH   19200
#define NCH     85          // 4 box + 1 conf + 80 cls
#define TILE    128         // anchors per block (= threads per block)
#define BPI     (NANCH / TILE)   // 150 blocks per image
#define NBLK    (BATCH * BPI)    // 4800
#define NV4     (TILE * NCH / 4) // 2720 16B chunks per tensor tile
#define EPSF    1e-7f
#define INV_PI2_4 0.40528473456935108577551785283891f  // 4/pi^2

typedef float v2f __attribute__((ext_vector_type(2)));
typedef float v8f __attribute__((ext_vector_type(8)));

// ---------------- CDNA5 async global -> LDS copy ----------------
__device__ __forceinline__ void async_copy_b128(uint32_t lds_byte_off, const void* gptr) {
    // VDST = VGPR with LDS byte address, VADDR = VGPR pair with 64-bit global addr,
    // SADDR = off (GV mode). Tracked by ASYNCcnt.
    asm volatile("global_load_async_to_lds_b128 %0, %1, off"
                 :: "v"(lds_byte_off), "v"((unsigned long long)(uintptr_t)gptr)
                 : "memory");
}

__device__ __forceinline__ void wait_async_zero() {
#if __has_builtin(__builtin_amdgcn_s_wait_asynccnt)
    __builtin_amdgcn_s_wait_asynccnt(0);
#else
    asm volatile("s_wait_asynccnt 0x0" ::: "memory");
#endif
}

// ---------------- wave32 sum using V_WMMA_F32_16X16X4_F32 ----------------
// A = ones(16x4), B vgpr0 = per-lane x, vgpr1 = 0  =>  D[m][n] = x_n + x_{n+16}.
// One WMMA folds 32 lanes -> 16 column sums (f32, no precision loss); finish
// with 4 shfl_xor folds (layout-permutation-invariant: every lane value enters
// exactly one B slot and all columns get folded).
__device__ __forceinline__ float wave_sum_f32(float x) {
#if __has_builtin(__builtin_amdgcn_wmma_f32_16x16x4_f32)
    v2f a; a[0] = 1.0f; a[1] = 1.0f;
    v2f b; b[0] = x;    b[1] = 0.0f;
    v8f c = {};
    c = __builtin_amdgcn_wmma_f32_16x16x4_f32(false, a, false, b, (short)0, c, false, false);
    x = c[0];
    x += __shfl_xor(x, 8, 32);
    x += __shfl_xor(x, 4, 32);
    x += __shfl_xor(x, 2, 32);
    x += __shfl_xor(x, 1, 32);
#else
    for (int off = 16; off >= 1; off >>= 1) x += __shfl_xor(x, off, 32);
#endif
    return x;
}

// softplus-style BCE-with-logits pieces (f32, matches jnp formulation)
__device__ __forceinline__ float bce_logits(float x, float t) {
    return fmaxf(x, 0.0f) - x * t + log1pf(expf(-fabsf(x)));
}

// ---------------- kernel 1: per-block partial sums ----------------
// ws[blk*5 + {0..4}] = { sum ciou*posf, sum cls_mean*posf, sum obj,
//                        sum pos_obj*posf, sum posf }
__global__ __launch_bounds__(TILE)
void ObjectDetectionLoss_partials(const float* __restrict__ outs,
                                  const float* __restrict__ tgts,
                                  float* __restrict__ ws) {
    __shared__ __align__(16) float s_o[TILE * NCH];
    __shared__ __align__(16) float s_t[TILE * NCH];
    __shared__ float s_wave[(TILE / 32) * 5];

    const int tid     = threadIdx.x;
    const int img     = blockIdx.x / BPI;
    const int tileIdx = blockIdx.x % BPI;
    const size_t base = ((size_t)img * NANCH + (size_t)tileIdx * TILE) * NCH; // float idx, 16B aligned

    // ---- stage both tiles into LDS via async b128 copies (coalesced) ----
    const char* go = (const char*)(outs + base);
    const char* gt = (const char*)(tgts + base);
    const uint32_t lo = (uint32_t)(uintptr_t)&s_o[0];
    const uint32_t lt = (uint32_t)(uintptr_t)&s_t[0];
    for (int i = tid; i < NV4; i += TILE) {
        async_copy_b128(lo + (uint32_t)i * 16u, go + (size_t)i * 16);
        async_copy_b128(lt + (uint32_t)i * 16u, gt + (size_t)i * 16);
    }
    wait_async_zero();
    __syncthreads();

    // ---- per-anchor math (one anchor per thread, stride-85 LDS: conflict-free) ----
    const float* o = &s_o[tid * NCH];
    const float* t = &s_t[tid * NCH];

    const float pcx = o[0], pcy = o[1], pw = o[2], ph = o[3], pconf = o[4];
    const float gcx = t[0], gcy = t[1], gw = t[2], gh = t[3], gconf = t[4];

    const bool  pos  = gconf > 0.5f;
    const float posf = pos ? 1.0f : 0.0f;

    // CIoU
    const float px1 = pcx - pw * 0.5f, py1 = pcy - ph * 0.5f;
    const float px2 = pcx + pw * 0.5f, py2 = pcy + ph * 0.5f;
    const float gx1 = gcx - gw * 0.5f, gy1 = gcy - gh * 0.5f;
    const float gx2 = gcx + gw * 0.5f, gy2 = gcy + gh * 0.5f;
    const float iw = fmaxf(fminf(px2, gx2) - fmaxf(px1, gx1), 0.0f);
    const float ih = fmaxf(fminf(py2, gy2) - fmaxf(py1, gy1), 0.0f);
    const float inter = iw * ih;
    const float wp = px2 - px1, hp = py2 - py1;
    const float wg = gx2 - gx1, hg = gy2 - gy1;
    const float uni  = wp * hp + wg * hg - inter;
    const float iou  = inter / (uni + EPSF);
    const float cw = fmaxf(px2, gx2) - fminf(px1, gx1);
    const float chh = fmaxf(py2, gy2) - fminf(py1, gy1);
    const float diag = cw * cw + chh * chh + EPSF;
    const float dx = px1 + px2 - gx1 - gx2;
    const float dy = py1 + py2 - gy1 - gy2;
    const float dist = (dx * dx + dy * dy) * 0.25f;
    const float dv = atanf(wg / (hg + EPSF)) - atanf(wp / (hp + EPSF));
    const float v  = INV_PI2_4 * dv * dv;
    const float aa = v / (1.0f - iou + v + EPSF);
    const float ciou = 1.0f - iou + dist / diag + aa * v;

    // class focal: alpha=0.5, gamma=0  =>  0.5 * bce, mean over 80 channels
    float clsum = 0.0f;
#pragma unroll 4
    for (int c = 5; c < NCH; ++c)
        clsum += bce_logits(o[c], t[c]);
    const float cls_mean = 0.5f * clsum * (1.0f / 80.0f);

    // objectness
    const float x   = pconf;
    const float sp  = log1pf(expf(-fabsf(x)));
    const float ce1 = fmaxf(x, 0.0f) - x + sp;            // bce(x, 1)
    const float ce0 = fmaxf(x, 0.0f) + sp;                // bce(x, 0)
    const float p   = 1.0f / (1.0f + expf(-x));
    const float pos_obj = 0.75f * ce1;                    // 1.5 (POS_W) * 0.5 (alpha_t)
    const float neg_obj = 0.25f * ce0 * (p * sqrtf(p));   // 0.5 (NEG_W) * 0.5 * p^1.5
    const float obj = pos ? pos_obj : neg_obj;

    // ---- wave-level f32 reduction via WMMA, then cross-wave via LDS ----
    const float r0 = wave_sum_f32(posf * ciou);
    const float r1 = wave_sum_f32(posf * cls_mean);
    const float r2 = wave_sum_f32(obj);
    const float r3 = wave_sum_f32(posf * pos_obj);
    const float r4 = wave_sum_f32(posf);

    if ((tid & 31) == 0) {
        const int w = tid >> 5;
        s_wave[w * 5 + 0] = r0; s_wave[w * 5 + 1] = r1; s_wave[w * 5 + 2] = r2;
        s_wave[w * 5 + 3] = r3; s_wave[w * 5 + 4] = r4;
    }
    __syncthreads();
    if (tid == 0) {
        float a0 = 0, a1 = 0, a2 = 0, a3 = 0, a4 = 0;
#pragma unroll
        for (int w = 0; w < TILE / 32; ++w) {
            a0 += s_wave[w * 5 + 0]; a1 += s_wave[w * 5 + 1]; a2 += s_wave[w * 5 + 2];
            a3 += s_wave[w * 5 + 3]; a4 += s_wave[w * 5 + 4];
        }
        float* p5 = ws + (size_t)blockIdx.x * 5;
        p5[0] = a0; p5[1] = a1; p5[2] = a2; p5[3] = a3; p5[4] = a4;
    }
}

// ---------------- kernel 2: deterministic finish ----------------
// out = { total, 0.5*avg_ciou, 1.5*avg_obj, 1.5*avg_cls, avg_pos_obj }
__global__ __launch_bounds__(256)
void ObjectDetectionLoss_finish(const float* __restrict__ ws, float* __restrict__ out) {
    __shared__ float s_img[BATCH][4];
    const int t   = threadIdx.x;
    const int img = t >> 3;          // 32 images, 8 threads each
    const int j   = t & 7;

    float s0 = 0, s1 = 0, s2 = 0, s3 = 0, s4 = 0;
    for (int k = j; k < BPI; k += 8) {
        const float* p = ws + ((size_t)img * BPI + k) * 5;
        s0 += p[0]; s1 += p[1]; s2 += p[2]; s3 += p[3]; s4 += p[4];
    }
#pragma unroll
    for (int off = 1; off < 8; off <<= 1) {   // fold 8-lane group (stays in group)
        s0 += __shfl_xor(s0, off, 32);
        s1 += __shfl_xor(s1, off, 32);
        s2 += __shfl_xor(s2, off, 32);
        s3 += __shfl_xor(s3, off, 32);
        s4 += __shfl_xor(s4, off, 32);
    }
    if (j == 0) {
        s_img[img][0] = s0;                       // sum ciou*posf
        s_img[img][1] = s1;                       // sum cls*posf
        s_img[img][2] = s2;                       // sum obj
        s_img[img][3] = s3 / fmaxf(s4, 1.0f);     // pos_obj_img
    }
    __syncthreads();
    if (t == 0) {
        float c = 0, l = 0, ob = 0, po = 0;
#pragma unroll
        for (int b = 0; b < BATCH; ++b) {
            c += s_img[b][0]; l += s_img[b][1]; ob += s_img[b][2]; po += s_img[b][3];
        }
        const float invBN = 1.0f / ((float)BATCH * (float)NANCH);
        const float o1 = 0.5f * (c  * invBN);   // BOX_W * avg_ciou
        const float o2 = 1.5f * (ob * invBN);   // OBJ_W * avg_obj
        const float o3 = 1.5f * (l  * invBN);   // CLS_W * avg_cls
        out[0] = o1 + o2 + o3;
        out[1] = o1;
        out[2] = o2;
        out[3] = o3;
        out[4] = po / (float)BATCH;             // avg_pos_obj
    }
}

// ---------------- launcher ----------------
extern "C" void kernel_launch(void* const* d_in, const int* in_sizes, int n_in,
                              void* d_out, int out_size, void* d_ws, size_t ws_size,
                              hipStream_t stream) {
    const float* outs = (const float*)d_in[0];
    const float* tgts = (const float*)d_in[1];
    float* ws  = (float*)d_ws;     // needs NBLK*5*4 = 96000 bytes
    float* out = (float*)d_out;    // 5 floats
    (void)in_sizes; (void)n_in; (void)out_size; (void)ws_size;

    ObjectDetectionLoss_partials<<<NBLK, TILE, 0, stream>>>(outs, tgts, ws);
    ObjectDetectionLoss_finish<<<1, 256, 0, stream>>>(ws, out);
}